// model_31284541784295
// MI455X (gfx1250) — compile-verified
//
#include <hip/hip_runtime.h>
#include <hip/hip_bf16.h>
#include <math.h>

// ---------------- problem constants ----------------
#define S_LEN 1024
#define VOCAB 32000
#define HID   1024
#define G4H   4096      // 4*HID
#define NWG_REC 16      // workgroups cooperating on the recurrences

typedef __attribute__((ext_vector_type(2))) float v2f;
typedef __attribute__((ext_vector_type(8))) float v8f;

// ---------------- small helpers ----------------
__device__ __forceinline__ float sigmoidf_(float x) {
    return 1.0f / (1.0f + __expf(-x));
}

// Device-wide generation barrier (persistent-grid, graph-capture safe).
__device__ __forceinline__ void grid_barrier(unsigned* cnt, unsigned* gen, unsigned nwg) {
    __syncthreads();
    if (threadIdx.x == 0) {
        unsigned g = __hip_atomic_load(gen, __ATOMIC_ACQUIRE, __HIP_MEMORY_SCOPE_AGENT);
        unsigned a = __hip_atomic_fetch_add(cnt, 1u, __ATOMIC_ACQ_REL, __HIP_MEMORY_SCOPE_AGENT);
        if (a == nwg - 1u) {
            __hip_atomic_store(cnt, 0u, __ATOMIC_RELAXED, __HIP_MEMORY_SCOPE_AGENT);
            __hip_atomic_fetch_add(gen, 1u, __ATOMIC_ACQ_REL, __HIP_MEMORY_SCOPE_AGENT);
        } else {
            while (__hip_atomic_load(gen, __ATOMIC_ACQUIRE, __HIP_MEMORY_SCOPE_AGENT) == g) { }
        }
    }
    __syncthreads();
}

// ---------------- init ----------------
__global__ void init_kernel(float* h_state, float* c_state, unsigned* bar) {
    for (int j = threadIdx.x; j < HID; j += blockDim.x) { h_state[j] = 0.f; c_state[j] = 0.f; }
    if (threadIdx.x < 2) bar[threadIdx.x] = 0u;
}

// ---------------- embedding gather (+ relu'd shifted gold) ----------------
__global__ void gather_kernel(const int* __restrict__ input_seq, const int* __restrict__ gold_seq,
                              const float* __restrict__ embed,
                              float* __restrict__ x_emb, float* __restrict__ prev_relu) {
    const int s = blockIdx.x;
    const int ri = input_seq[s];
    const int rp = (s == 0) ? -1 : gold_seq[s - 1];
    for (int j = threadIdx.x; j < HID; j += blockDim.x) {
        x_emb[(size_t)s * HID + j] = embed[(size_t)ri * HID + j];
        float pv = (rp < 0) ? 0.f : embed[(size_t)rp * HID + j];
        prev_relu[(size_t)s * HID + j] = fmaxf(pv, 0.f);
    }
}

// ---------------- transpose: W (R x C) -> WT (C x R) ----------------
__global__ void transpose_kernel(const float* __restrict__ W, float* __restrict__ WT,
                                 int R, int C) {
    __shared__ float tile[32][33];
    const int c0 = blockIdx.x * 32, r0 = blockIdx.y * 32;
    for (int i = threadIdx.y; i < 32; i += 8)
        tile[i][threadIdx.x] = W[(size_t)(r0 + i) * C + c0 + threadIdx.x];
    __syncthreads();
    for (int i = threadIdx.y; i < 32; i += 8)
        WT[(size_t)(c0 + i) * R + r0 + threadIdx.x] = tile[threadIdx.x][i];
}

// ---------------- fp32 WMMA GEMM:  C[M,N] = A[M,K] * B[N,K]^T (+bias[N]) ----------------
// Tiles: block = 256 threads (8 waves) -> 128x128 of C; K staged 32 at a time in LDS.
// Uses V_WMMA_F32_16X16X4_F32 (fp32 in/out, matches reference precision exactly).
#define BM 128
#define BN 128
#define KC 32

__global__ __launch_bounds__(256)
void wmma_gemm_nt(const float* __restrict__ A, int lda,
                  const float* __restrict__ B, int ldb,
                  const float* __restrict__ bias,
                  float* __restrict__ C, int ldc, int K) {
    __shared__ float As[BM][KC + 1];
    __shared__ float Bs[BN][KC + 1];

    const int tid  = threadIdx.x;
    const int wave = tid >> 5;          // 0..7
    const int lane = tid & 31;
    const int m0   = blockIdx.y * BM;
    const int n0   = blockIdx.x * BN;

    v8f acc[8];
    #pragma unroll
    for (int i = 0; i < 8; ++i) acc[i] = v8f{0.f,0.f,0.f,0.f,0.f,0.f,0.f,0.f};

    for (int k0 = 0; k0 < K; k0 += KC) {
        // stage A/B chunks into LDS: 4096 floats each, 4x float4 per thread
        #pragma unroll
        for (int r = 0; r < 4; ++r) {
            int slot = tid + r * 256;       // 1024 float4 slots
            int row  = slot >> 3;           // 0..127
            int col4 = slot & 7;            // 0..7
            const float4 av = *reinterpret_cast<const float4*>(
                &A[(size_t)(m0 + row) * lda + k0 + col4 * 4]);
            As[row][col4 * 4 + 0] = av.x; As[row][col4 * 4 + 1] = av.y;
            As[row][col4 * 4 + 2] = av.z; As[row][col4 * 4 + 3] = av.w;
            const float4 bv = *reinterpret_cast<const float4*>(
                &B[(size_t)(n0 + row) * ldb + k0 + col4 * 4]);
            Bs[row][col4 * 4 + 0] = bv.x; Bs[row][col4 * 4 + 1] = bv.y;
            Bs[row][col4 * 4 + 2] = bv.z; Bs[row][col4 * 4 + 3] = bv.w;
        }
        if (k0 + KC < K) {  // pull next chunk toward L2 (global_prefetch_b8)
            __builtin_prefetch(&A[(size_t)(m0 + (tid & 127)) * lda + k0 + KC], 0, 1);
            __builtin_prefetch(&B[(size_t)(n0 + (tid & 127)) * ldb + k0 + KC], 0, 1);
        }
        __syncthreads();

        #pragma unroll
        for (int kk = 0; kk < KC; kk += 4) {
            // A frag (16x4 f32): lanes 0-15 hold K=kk,kk+1; lanes 16-31 hold K=kk+2,kk+3
            const int ar = (wave << 4) + (lane & 15);
            const int ak = kk + ((lane >> 4) << 1);
            v2f afrag; afrag.x = As[ar][ak]; afrag.y = As[ar][ak + 1];
            #pragma unroll
            for (int nt = 0; nt < 8; ++nt) {
                const int bc = (nt << 4) + (lane & 15);
                v2f bfrag; bfrag.x = Bs[bc][ak]; bfrag.y = Bs[bc][ak + 1];
                acc[nt] = __builtin_amdgcn_wmma_f32_16x16x4_f32(
                    false, afrag, false, bfrag, (short)0, acc[nt], false, false);
            }
        }
        __syncthreads();
    }

    // epilogue: C/D layout — VGPR v: lanes0-15 -> M=v, lanes16-31 -> M=v+8
    const int mbase = m0 + (wave << 4) + ((lane >> 4) << 3);
    #pragma unroll
    for (int nt = 0; nt < 8; ++nt) {
        const int n = n0 + (nt << 4) + (lane & 15);
        const float bv = bias ? bias[n] : 0.f;
        #pragma unroll
        for (int v = 0; v < 8; ++v)
            C[(size_t)(mbase + v) * ldc + n] = acc[nt][v] + bv;
    }
}

// ---------------- ctxpre[r] = dec_b[r] + dec_Wih[r, H: ] . context ----------------
__global__ void ctxpre_kernel(const float* __restrict__ dec_Wih, const float* __restrict__ dec_b,
                              const float* __restrict__ context, float* __restrict__ ctxpre) {
    const int r = blockIdx.x * blockDim.x + threadIdx.x;
    if (r >= G4H) return;
    const float* w = dec_Wih + (size_t)r * (2 * HID) + HID;
    float s = dec_b[r];
    for (int k = 0; k < HID; ++k) s = fmaf(w[k], context[k], s);
    ctxpre[r] = s;
}

// ---------------- LSTM recurrence over 16 cooperating workgroups ----------------
// gate_base[t] already holds the h-independent part (input GEMM + bias terms).
// WhhT is (HID x G4H): WhhT[k*G4H + gate*HID + cell].  Each step:
//   gates = gate_base[t] + WhhT^T h ; standard LSTM update; h published via L2 + barrier.
template<bool IS_ENCODER>
__global__ __launch_bounds__(1024)
void lstm_recurrence(const float* __restrict__ gate_base,  // S x 4H
                     const float* __restrict__ extra,      // 4H (decoder ctxpre) or null
                     const float* __restrict__ WhhT,       // HID x 4H
                     float* __restrict__ h_seq,            // S x HID output (enc_out / Hdec)
                     float* __restrict__ h_state, float* __restrict__ c_state,
                     float* __restrict__ context,          // HID (encoder only)
                     unsigned* bar) {
    const int tid  = threadIdx.x;
    const int wg   = blockIdx.x;            // 0..15
    const int part = tid >> 6;              // 0..15 : K chunk of 64
    const int cidl = tid & 63;              // 0..63 : local cell
    const int cell = wg * 64 + cidl;        // global cell this block owns
    const int kbeg = part * 64;

    __shared__ float hsh[HID];
    __shared__ float partial[16 * 256];     // [part][cidl*4 + gate]

    hsh[tid] = h_state[tid];
    float c_reg = 0.f, ctx_acc = 0.f;
    if (tid < 64) c_reg = c_state[wg * 64 + tid];
    __syncthreads();

    for (int t = 0; t < S_LEN; ++t) {
        float p0 = 0.f, p1 = 0.f, p2 = 0.f, p3 = 0.f;
        #pragma unroll 4
        for (int j = 0; j < 64; ++j) {
            const float hv = hsh[kbeg + j];                       // LDS broadcast
            const float* w = WhhT + (size_t)(kbeg + j) * G4H + cell;  // coalesced over lanes
            p0 = fmaf(hv, w[0],        p0);
            p1 = fmaf(hv, w[HID],      p1);
            p2 = fmaf(hv, w[2 * HID],  p2);
            p3 = fmaf(hv, w[3 * HID],  p3);
        }
        const int pb = part * 256 + cidl * 4;
        partial[pb + 0] = p0; partial[pb + 1] = p1;
        partial[pb + 2] = p2; partial[pb + 3] = p3;
        __syncthreads();

        if (tid < 64) {
            const int cg = wg * 64 + tid;
            float g0 = 0.f, g1 = 0.f, g2 = 0.f, g3 = 0.f;
            #pragma unroll
            for (int p = 0; p < 16; ++p) {
                const int b = p * 256 + tid * 4;
                g0 += partial[b + 0]; g1 += partial[b + 1];
                g2 += partial[b + 2]; g3 += partial[b + 3];
            }
            const size_t tb = (size_t)t * G4H;
            g0 += gate_base[tb + cg];
            g1 += gate_base[tb + HID + cg];
            g2 += gate_base[tb + 2 * HID + cg];
            g3 += gate_base[tb + 3 * HID + cg];
            if (!IS_ENCODER) {
                g0 += extra[cg];            g1 += extra[HID + cg];
                g2 += extra[2 * HID + cg];  g3 += extra[3 * HID + cg];
            }
            const float i  = sigmoidf_(g0);
            const float f  = sigmoidf_(g1);
            const float gg = tanhf(g2);
            const float o  = sigmoidf_(g3);
            c_reg = f * c_reg + i * gg;
            const float hn = o * tanhf(c_reg);
            h_state[cg] = hn;                          // publish via L2
            h_seq[(size_t)t * HID + cg] = hn;
            if (IS_ENCODER) ctx_acc += hn;
        }
        __threadfence();
        grid_barrier(bar, bar + 1, NWG_REC);
        hsh[tid] = h_state[tid];                       // refresh full h
        __syncthreads();
    }

    if (tid < 64) {
        const int cg = wg * 64 + tid;
        c_state[cg] = c_reg;
        if (IS_ENCODER) context[cg] = ctx_acc;
    }
}

// ---------------- per-row argmax over VOCAB ----------------
__global__ void argmax_kernel(const float* __restrict__ logits, float* __restrict__ idx_out) {
    __shared__ float sv[256];
    __shared__ int   si[256];
    const int s = blockIdx.x;
    const float* row = logits + (size_t)s * VOCAB;
    float best = -INFINITY; int bi = 0;
    for (int j = threadIdx.x; j < VOCAB; j += 256) {
        const float v = row[j];
        if (v > best) { best = v; bi = j; }
    }
    sv[threadIdx.x] = best; si[threadIdx.x] = bi;
    __syncthreads();
    for (int off = 128; off > 0; off >>= 1) {
        if (threadIdx.x < off) {
            const float v2 = sv[threadIdx.x + off]; const int i2 = si[threadIdx.x + off];
            if (v2 > sv[threadIdx.x] || (v2 == sv[threadIdx.x] && i2 < si[threadIdx.x])) {
                sv[threadIdx.x] = v2; si[threadIdx.x] = i2;
            }
        }
        __syncthreads();
    }
    if (threadIdx.x == 0) idx_out[s] = (float)si[0];
}

// ---------------- driver ----------------
extern "C" void kernel_launch(void* const* d_in, const int* in_sizes, int n_in,
                              void* d_out, int out_size, void* d_ws, size_t ws_size,
                              hipStream_t stream) {
    const int*   input_seq = (const int*)d_in[0];
    const int*   gold_seq  = (const int*)d_in[1];
    const float* embed     = (const float*)d_in[2];
    const float* enc_Wih   = (const float*)d_in[3];
    const float* enc_Whh   = (const float*)d_in[4];
    const float* enc_b     = (const float*)d_in[5];
    const float* dec_Wih   = (const float*)d_in[6];
    const float* dec_Whh   = (const float*)d_in[7];
    const float* dec_b     = (const float*)d_in[8];
    // d_in[9..14]: attention params — provably dead (softmax over (S,1) axis=1 == ones)
    const float* out_w     = (const float*)d_in[15];
    const float* out_b     = (const float*)d_in[16];

    float* preds = (float*)d_out;                       // S x V logits
    float* idxs  = preds + (size_t)S_LEN * VOCAB;       // S predicted ids (as float)

    // workspace carve (floats)
    float* ws = (float*)d_ws;
    size_t off = 0;
    float* x_emb   = ws + off; off += (size_t)S_LEN * HID;
    float* prevr   = ws + off; off += (size_t)S_LEN * HID;
    float* Xpre    = ws + off; off += (size_t)S_LEN * G4H;
    float* Ppre    = ws + off; off += (size_t)S_LEN * G4H;
    float* enc_out = ws + off; off += (size_t)S_LEN * HID;
    float* Hdec    = ws + off; off += (size_t)S_LEN * HID;
    float* WhhTe   = ws + off; off += (size_t)HID * G4H;
    float* WhhTd   = ws + off; off += (size_t)HID * G4H;
    float* h_state = ws + off; off += HID;
    float* c_state = ws + off; off += HID;
    float* context = ws + off; off += HID;
    float* ctxpre  = ws + off; off += G4H;
    unsigned* bar  = (unsigned*)(ws + off);

    init_kernel<<<1, 256, 0, stream>>>(h_state, c_state, bar);
    gather_kernel<<<S_LEN, 256, 0, stream>>>(input_seq, gold_seq, embed, x_emb, prevr);

    dim3 tb(32, 8);
    transpose_kernel<<<dim3(HID / 32, G4H / 32), tb, 0, stream>>>(enc_Whh, WhhTe, G4H, HID);
    transpose_kernel<<<dim3(HID / 32, G4H / 32), tb, 0, stream>>>(dec_Whh, WhhTd, G4H, HID);

    // Xpre = x_emb @ enc_Wih^T + enc_b          (1024 x 4096, K=1024)
    wmma_gemm_nt<<<dim3(G4H / BN, S_LEN / BM), 256, 0, stream>>>(
        x_emb, HID, enc_Wih, HID, enc_b, Xpre, G4H, HID);
    // Ppre = relu(prev) @ dec_Wih[:, :H]^T      (bias folded into ctxpre)
    wmma_gemm_nt<<<dim3(G4H / BN, S_LEN / BM), 256, 0, stream>>>(
        prevr, HID, dec_Wih, 2 * HID, nullptr, Ppre, G4H, HID);

    lstm_recurrence<true><<<NWG_REC, 1024, 0, stream>>>(
        Xpre, nullptr, WhhTe, enc_out, h_state, c_state, context, bar);

    ctxpre_kernel<<<G4H / 256, 256, 0, stream>>>(dec_Wih, dec_b, context, ctxpre);

    lstm_recurrence<false><<<NWG_REC, 1024, 0, stream>>>(
        Ppre, ctxpre, WhhTd, Hdec, h_state, c_state, nullptr, bar);

    // logits = Hdec @ out_w^T + out_b           (1024 x 32000, K=1024)
    wmma_gemm_nt<<<dim3(VOCAB / BN, S_LEN / BM), 256, 0, stream>>>(
        Hdec, HID, out_w, HID, out_b, preds, VOCAB, HID);

    argmax_kernel<<<S_LEN, 256, 0, stream>>>(preds, idxs);
}